// InstanceNorm_25340307046985
// MI455X (gfx1250) — compile-verified
//
#include <hip/hip_runtime.h>
#include <stdint.h>

// irreps = 128x0e + 64x1e + 32x2e
#define DIM      480
#define NSCALAR  128
#define NFEAT    224
#define NQ       352      // 128 scalar values + 224 per-channel sum-squares
#define NGRAPH   512
#define TILE     16
#define EPS      1e-5f

typedef float v2f __attribute__((ext_vector_type(2)));
typedef float v4f __attribute__((ext_vector_type(4)));
typedef float v8f __attribute__((ext_vector_type(8)));

// -------- Pass 1: segmented per-graph sums via f32 WMMA + async LDS staging ----
// Block = 256 threads (8 waves) owns a 16-node tile (16 contiguous rows = 30KB).
// Tile is staged to LDS with GLOBAL_LOAD_ASYNC_TO_LDS_B128 (ASYNCcnt).
// A-matrix (16x4 f32) encodes node -> distinct-graph-slot membership; B holds
// 4 nodes x 16 channels of the per-node stats vector q:
//   q[0..127]   scalar values, q[128..255] scalar squares,
//   q[256..319] l=1 sum of 3 squares, q[320..351] l=2 sum of 5 squares.
__global__ void __launch_bounds__(256)
stats_kernel(const float* __restrict__ inp, const int* __restrict__ batch,
             float* __restrict__ accum, float* __restrict__ cnt, int N)
{
  __shared__ __align__(16) float s_data[TILE * DIM];   // 30720 B
  __shared__ int s_batch[TILE];
  __shared__ int s_rank[TILE];
  __shared__ int s_gid[TILE];
  __shared__ int s_nd;

  const int tileBase = blockIdx.x * TILE;
  const int tid  = threadIdx.x;
  const int wave = tid >> 5;
  const int lane = tid & 31;

  const int nvalid = min(TILE, N - tileBase);
  const int nchunk = (nvalid * DIM) >> 2;              // 16-byte chunks

  // Async-stage the contiguous tile into LDS (each byte read from HBM once).
  {
    const float* gbase = inp + (size_t)tileBase * DIM;
    for (int i = tid; i < nchunk; i += 256) {
      unsigned lds_off = (unsigned)(uintptr_t)(&s_data[i * 4]);
      unsigned long long gaddr = (unsigned long long)(uintptr_t)(gbase + i * 4);
      asm volatile("global_load_async_to_lds_b128 %0, %1, off"
                   :: "v"(lds_off), "v"(gaddr) : "memory");
    }
  }

  if (tid < TILE) {
    int node = tileBase + tid;
    s_batch[tid] = (node < N) ? batch[node] : -1;
    s_gid[tid]   = -1;
  }
  __syncthreads();
  if (tid < TILE) {
    int r = 0;
    for (int j = 1; j <= tid; ++j) r += (s_batch[j] != s_batch[j - 1]);
    s_rank[tid] = r;
    if (s_batch[tid] >= 0) atomicAdd(&cnt[s_batch[tid]], 1.0f);   // node counts
  }
  __syncthreads();
  if (tid < TILE) {
    if (tid == 0 || s_rank[tid] != s_rank[tid - 1]) s_gid[s_rank[tid]] = s_batch[tid];
    if (tid == TILE - 1) s_nd = s_rank[tid] + 1;
  }
  // Our own async copies must be complete before the barrier makes LDS visible.
  asm volatile("s_wait_asynccnt 0x0" ::: "memory");
  __syncthreads();

  const int khalf = lane >> 4;     // lanes 16-31: second K pair / D rows 8-15
  const int nloc  = lane & 15;

  // A-matrix (16x4 f32): lane m supplies A[m][2*khalf + {0,1}] per K-step.
  v2f a[4];
  const int gm = s_gid[nloc];
#pragma unroll
  for (int c = 0; c < 4; ++c) {
    int k0 = 4 * c + 2 * khalf;
    a[c].x = (gm >= 0 && s_batch[k0 + 0] == gm) ? 1.0f : 0.0f;
    a[c].y = (gm >= 0 && s_batch[k0 + 1] == gm) ? 1.0f : 0.0f;
  }
  const int nd = s_nd;

  for (int cg = wave; cg < (NQ / TILE); cg += 8) {
    // cg is wave-uniform: scalarize so section select is s_cbranch, not exec mask.
    const int cgu = __builtin_amdgcn_readfirstlane(cg);
    v8f acc = {};
#pragma unroll
    for (int c = 0; c < 4; ++c) {
      const int kp = 4 * c + 2 * khalf;      // local node pair base
      v2f b;
      if (cgu < 8) {                          // scalar values
        int col = cgu * 16 + nloc;
        b.x = s_data[(kp + 0) * DIM + col];
        b.y = s_data[(kp + 1) * DIM + col];
      } else if (cgu < 16) {                  // scalar squares
        int col = (cgu - 8) * 16 + nloc;
        float x0 = s_data[(kp + 0) * DIM + col];
        float x1 = s_data[(kp + 1) * DIM + col];
        b.x = x0 * x0; b.y = x1 * x1;
      } else if (cgu < 20) {                  // l=1: 3 squares
        int ch = (cgu - 16) * 16 + nloc;
        const float* p0 = &s_data[(kp + 0) * DIM + 128 + ch * 3];
        const float* p1 = &s_data[(kp + 1) * DIM + 128 + ch * 3];
        b.x = p0[0]*p0[0] + p0[1]*p0[1] + p0[2]*p0[2];
        b.y = p1[0]*p1[0] + p1[1]*p1[1] + p1[2]*p1[2];
      } else {                                // l=2: 5 squares
        int ch = (cgu - 20) * 16 + nloc;
        const float* p0 = &s_data[(kp + 0) * DIM + 320 + ch * 5];
        const float* p1 = &s_data[(kp + 1) * DIM + 320 + ch * 5];
        b.x = p0[0]*p0[0]+p0[1]*p0[1]+p0[2]*p0[2]+p0[3]*p0[3]+p0[4]*p0[4];
        b.y = p1[0]*p1[0]+p1[1]*p1[1]+p1[2]*p1[2]+p1[3]*p1[3]+p1[4]*p1[4];
      }
      if (kp + 0 >= nvalid) b.x = 0.0f;       // padded nodes contribute zero
      if (kp + 1 >= nvalid) b.y = 0.0f;
      // D(16x16,f32) = A(16x4) * B(4x16) + C  — exact f32 segmented partial sum
      acc = __builtin_amdgcn_wmma_f32_16x16x4_f32(false, a[c], false, b,
                                                  (short)0, acc, false, false);
    }
    const int qc0 = cgu * TILE;
    // D layout: VGPR v -> row (v + 8*khalf), col = lane&15. Flush non-empty rows.
#pragma unroll
    for (int v = 0; v < 8; ++v) {
      int row = v + 8 * khalf;
      if (row < nd) {
        int g = s_gid[row];
        if (g >= 0) atomicAdd(&accum[(size_t)g * NQ + qc0 + nloc], acc[v]);
      }
    }
  }
}

// -------- Pass 2: per-(graph, channel) normalizer --------------------------
__global__ void finalize_kernel(const float* __restrict__ accum,
                                const float* __restrict__ cnt,
                                const float* __restrict__ weight,
                                float* __restrict__ meanv,
                                float* __restrict__ scalev)
{
  int i = blockIdx.x * blockDim.x + threadIdx.x;
  if (i >= NGRAPH * NFEAT) return;
  int g = i / NFEAT;
  int c = i - g * NFEAT;
  float n = fmaxf(cnt[g], 1.0f);
  const float* a = accum + (size_t)g * NQ;
  float sumsq = a[NSCALAR + c];           // sum-of-squares slot is uniformly c+128
  float norm;
  if (c < NSCALAR) {
    float mu  = a[c] / n;
    float var = sumsq / n - mu * mu;      // E[(x-mu)^2] = E[x^2] - mu^2 (exact)
    norm = rsqrtf(var + EPS) * weight[c];
    meanv[(size_t)g * NSCALAR + c] = mu;
  } else {
    float d = (c < 192) ? 3.0f : 5.0f;
    norm = rsqrtf(sumsq / (n * d) + EPS) * weight[c];
  }
  scalev[(size_t)g * NFEAT + c] = norm;
}

// -------- Pass 3: apply. Streaming, non-temporal; tables stay L2-resident ----
__global__ void __launch_bounds__(256)
apply_kernel(const float* __restrict__ inp, const int* __restrict__ batch,
             const float* __restrict__ meanv, const float* __restrict__ scalev,
             const float* __restrict__ bias, float* __restrict__ out, int N)
{
  long long idx   = (long long)blockIdx.x * blockDim.x + threadIdx.x;
  long long total = (long long)N * (DIM / 4);
  if (idx >= total) return;
  int node = (int)(idx / (DIM / 4));
  int c4   = (int)(idx - (long long)node * (DIM / 4));
  int g    = batch[node];
  const v4f* ip = (const v4f*)(inp + (size_t)node * DIM);
  v4f x = __builtin_nontemporal_load(ip + c4);
  float v[4] = {x.x, x.y, x.z, x.w};
  const float* sc = scalev + (size_t)g * NFEAT;
  int col0 = c4 * 4;
  float r[4];
#pragma unroll
  for (int j = 0; j < 4; ++j) {
    int col = col0 + j;
    if (col < NSCALAR) {
      r[j] = (v[j] - meanv[(size_t)g * NSCALAR + col]) * sc[col] + bias[col];
    } else if (col < 320) {
      r[j] = v[j] * sc[128 + (col - 128) / 3];
    } else {
      r[j] = v[j] * sc[192 + (col - 320) / 5];
    }
  }
  v4f y = {r[0], r[1], r[2], r[3]};
  __builtin_nontemporal_store(y, ((v4f*)(out + (size_t)node * DIM)) + c4);
}

__global__ void zero_kernel(float* __restrict__ p, int n) {
  int i = blockIdx.x * blockDim.x + threadIdx.x;
  if (i < n) p[i] = 0.0f;
}

extern "C" void kernel_launch(void* const* d_in, const int* in_sizes, int n_in,
                              void* d_out, int out_size, void* d_ws, size_t ws_size,
                              hipStream_t stream) {
  const float* inp    = (const float*)d_in[0];
  const float* weight = (const float*)d_in[1];
  const float* bias   = (const float*)d_in[2];
  const int*   batch  = (const int*)d_in[3];
  float*       out    = (float*)d_out;
  const int N = in_sizes[0] / DIM;

  // workspace layout (floats): accum[G*NQ] | cnt[G] | mean[G*128] | scale[G*224]
  float* accum  = (float*)d_ws;
  float* cnt    = accum + (size_t)NGRAPH * NQ;
  float* meanv  = cnt + NGRAPH;
  float* scalev = meanv + (size_t)NGRAPH * NSCALAR;

  int zn = NGRAPH * NQ + NGRAPH;
  zero_kernel<<<(zn + 255) / 256, 256, 0, stream>>>(accum, zn);

  int tiles = (N + TILE - 1) / TILE;
  stats_kernel<<<tiles, 256, 0, stream>>>(inp, batch, accum, cnt, N);

  int fin = NGRAPH * NFEAT;
  finalize_kernel<<<(fin + 255) / 256, 256, 0, stream>>>(accum, cnt, weight, meanv, scalev);

  long long tot = (long long)N * (DIM / 4);
  int blocks = (int)((tot + 255) / 256);
  apply_kernel<<<blocks, 256, 0, stream>>>(inp, batch, meanv, scalev, bias, out, N);
}